// LossFunction_FixedEncoder_Reg_KL_23270132809776
// MI455X (gfx1250) — compile-verified
//
#include <hip/hip_runtime.h>
#include <hip/hip_bf16.h>

// ---------------------------------------------------------------------------
// Fused NCC(9^3 box) + grad-L2 + KLDiv loss for fixed (1,*,160,192,160) fp32.
//
// NCC: separable 9-tap box filter as banded f32 GEMMs on the matrix pipe:
// Out(16x16) = Band(16x24) x Data(24x16) via six V_WMMA_F32_16X16X4_F32
// accumulation steps, 5 fields (I, J, I^2, J^2, IJ) computed concurrently
// with 5 accumulators sharing one set of LDS reads. W-pass then H-pass
// through LDS. D-axis: register-resident running 9-plane window in the
// native WMMA C/D layout (lane owns cells m=r+8*hi, n=lane&15), with a
// 9-slot LDS ring holding per-plane contributions for the subtraction.
// One wave per workgroup, each owning a 16x16 HxW tile over a 20-deep D
// segment (960 WGs). Interior tiles stage global->LDS with
// GLOBAL_LOAD_ASYNC_TO_LDS_B32 (ASYNCcnt); boundary tiles zero-pad in VALU.
// ---------------------------------------------------------------------------

typedef float v2f __attribute__((ext_vector_type(2)));
typedef float v8f __attribute__((ext_vector_type(8)));

#define D_ 160
#define H_ 192
#define W_ 160
#define TILE 16
#define HALO 4
#define IN_ 24              // TILE + 2*HALO
#define SEGLEN 20           // D segment per workgroup
#define SEGS (D_ / SEGLEN)  // 8
#define WIN_SZ 729.0f

// ws layout (floats): [0]=cc_sum [1]=dy_sum [2]=dx_sum [3]=dz_sum [4]=kl_sum

__device__ __forceinline__ v8f wmma_f32(v2f a, v2f b, v8f c) {
  // D = A(16x4,f32) * B(4x16,f32) + C(16x16,f32)
  return __builtin_amdgcn_wmma_f32_16x16x4_f32(
      false, a, false, b, (short)0, c, false, false);
}

// Band(16x24): Band[m][k] = 1 iff k in [m, m+8]  (9-tap box, 'same' pad 4)
__device__ __forceinline__ float band(int m, int k) {
  return (k >= m && k <= m + 8) ? 1.0f : 0.0f;
}

__global__ void __launch_bounds__(32)
ncc_kernel(const float* __restrict__ y, const float* __restrict__ tgt,
           float* __restrict__ ws) {
  __shared__ float Ilds[IN_][IN_ + 1];       // +1 pad: dodge bank conflicts
  __shared__ float Jlds[IN_][IN_ + 1];
  __shared__ float wf5[5][IN_][TILE + 1];    // W-filtered, all 5 fields
  __shared__ float ring[9][5][TILE * TILE];  // per-plane HW-filtered values

  const int lane = threadIdx.x;              // wave32: one wave per workgroup
  int bid = blockIdx.x;
  const int wb = bid % (W_ / TILE); bid /= (W_ / TILE);
  const int hb = bid % (H_ / TILE); bid /= (H_ / TILE);
  const int sb = bid;
  const int h0 = hb * TILE, w0 = wb * TILE, d0 = sb * SEGLEN;

  const int m16 = lane & 15;                 // N (B/C/D) or M (A) index
  const int hi  = lane >> 4;                 // half-wave K split

  // block-uniform: interior HW tiles need no zero padding -> pure async path
  const bool interior = (h0 - HALO >= 0) && (h0 + TILE + HALO <= H_) &&
                        (w0 - HALO >= 0) && (w0 + TILE + HALO <= W_);

  // loop-invariant banded A fragments (0/1 entries) kept in VGPRs
  v2f afrag[6];
#pragma unroll
  for (int kc = 0; kc < 6; ++kc) {
    const int kb = kc * 4 + (hi << 1);
    afrag[kc].x = band(m16, kb);
    afrag[kc].y = band(m16, kb + 1);
  }

  // zero the ring once: first 8 partial-window steps subtract zeros
  for (int t = lane; t < 9 * 5 * TILE * TILE; t += 32)
    (&ring[0][0][0])[t] = 0.0f;
  asm volatile("s_wait_dscnt 0" ::: "memory");

  // running 9-plane window sums, register-resident, in WMMA C/D layout
  v8f sI = {}, sJ = {}, sII = {}, sJJ = {}, sIJ = {};
  float cc_acc = 0.0f;

  for (int zp = 0; zp < SEGLEN + 2 * HALO; ++zp) {
    const int pi = d0 - HALO + zp;           // input plane (global D index)
    const int slot = zp % 9;

    // new plane's HW-filtered contribution (zero if plane outside volume)
    v8f nI = {}, nJ = {}, nII = {}, nJJ = {}, nIJ = {};

    if (pi >= 0 && pi < D_) {
      // ---- stage 24x24 I/J tiles of plane pi into LDS --------------------
      if (interior) {
        // async global->LDS DMA, no VGPR data movement (ASYNCcnt path)
        for (int t = lane; t < IN_ * IN_; t += 32) {
          const int r = t / IN_, c = t % IN_;
          const size_t g =
              ((size_t)pi * H_ + (h0 - HALO + r)) * W_ + (w0 - HALO + c);
          // low 32 bits of a generic pointer to a __shared__ object are the
          // workgroup-relative LDS byte offset
          const unsigned ioff =
              (unsigned)(unsigned long long)(const void*)&Ilds[r][c];
          const unsigned joff =
              (unsigned)(unsigned long long)(const void*)&Jlds[r][c];
          const unsigned long long ia =
              (unsigned long long)(const void*)(tgt + g);
          const unsigned long long ja =
              (unsigned long long)(const void*)(y + g);
          asm volatile("global_load_async_to_lds_b32 %0, %1, off"
                       :: "v"(ioff), "v"(ia) : "memory");
          asm volatile("global_load_async_to_lds_b32 %0, %1, off"
                       :: "v"(joff), "v"(ja) : "memory");
        }
        asm volatile("s_wait_asynccnt 0" ::: "memory");
      } else {
        // boundary tile: zero-padded scalar staging
        for (int t = lane; t < IN_ * IN_; t += 32) {
          const int r = t / IN_, c = t % IN_;
          const int gh = h0 - HALO + r, gw = w0 - HALO + c;
          float iv = 0.0f, jv = 0.0f;
          if (gh >= 0 && gh < H_ && gw >= 0 && gw < W_) {
            const size_t g = ((size_t)pi * H_ + gh) * W_ + gw;
            iv = tgt[g];                     // I = tgt (_nas_ncc(tgt, y))
            jv = y[g];                       // J = y
          }
          Ilds[r][c] = iv;
          Jlds[r][c] = jv;
        }
        asm volatile("s_wait_dscnt 0" ::: "memory");
      }
      // prefetch next plane's tile rows while we crunch this one
      if (pi + 1 < D_ && lane < IN_) {
        int gh = h0 - HALO + lane;
        gh = gh < 0 ? 0 : (gh >= H_ ? H_ - 1 : gh);
        const int gw = (w0 - HALO) < 0 ? 0 : (w0 - HALO);
        const size_t g = ((size_t)(pi + 1) * H_ + gh) * W_ + gw;
        __builtin_prefetch(&tgt[g], 0, 1);   // -> global_prefetch_b8
        __builtin_prefetch(&y[g],   0, 1);
      }

      // ---- pass 1: 9-tap box along W, 5 fields concurrently --------------
      // Out[m=w_out][n=h_row] = sum_k Band[m][k] * P[h_row][w_in=k]
#pragma unroll
      for (int g = 0; g < 2; ++g) {          // H-row groups 0-15, 16-23(+pad)
        v8f aI = {}, aJ = {}, aII = {}, aJJ = {}, aIJ = {};
        const int row = g * 16 + m16;        // B column n = m16
        const bool rok = (row < IN_);
#pragma unroll
        for (int kc = 0; kc < 6; ++kc) {
          const int kb = kc * 4 + (hi << 1); // lanes 16-31 supply K +2,+3
          float i0 = 0.f, i1 = 0.f, j0 = 0.f, j1 = 0.f;
          if (rok) {
            i0 = Ilds[row][kb]; i1 = Ilds[row][kb + 1];
            j0 = Jlds[row][kb]; j1 = Jlds[row][kb + 1];
          }
          const v2f bI  = {i0, i1};
          const v2f bJ  = {j0, j1};
          const v2f bII = {i0 * i0, i1 * i1};
          const v2f bJJ = {j0 * j0, j1 * j1};
          const v2f bIJ = {i0 * j0, i1 * j1};
          aI  = wmma_f32(afrag[kc], bI,  aI);
          aJ  = wmma_f32(afrag[kc], bJ,  aJ);
          aII = wmma_f32(afrag[kc], bII, aII);
          aJJ = wmma_f32(afrag[kc], bJJ, aJJ);
          aIJ = wmma_f32(afrag[kc], bIJ, aIJ);
        }
        // D VGPR r: M = r + 8*hi (= w_out), N = m16 (= h_row in group)
        if (rok) {
#pragma unroll
          for (int r = 0; r < 8; ++r) {
            const int m = r + hi * 8;
            wf5[0][row][m] = aI[r];
            wf5[1][row][m] = aJ[r];
            wf5[2][row][m] = aII[r];
            wf5[3][row][m] = aJJ[r];
            wf5[4][row][m] = aIJ[r];
          }
        }
      }
      asm volatile("s_wait_dscnt 0" ::: "memory");

      // ---- pass 2: 9-tap box along H, 5 fields concurrently --------------
      // Out[m=h_out][n=w_out] = sum_k Band[m][k] * wf[h_row=k][w_out=n]
#pragma unroll
      for (int kc = 0; kc < 6; ++kc) {
        const int kb = kc * 4 + (hi << 1);
        const v2f b0 = {wf5[0][kb][m16], wf5[0][kb + 1][m16]};
        const v2f b1 = {wf5[1][kb][m16], wf5[1][kb + 1][m16]};
        const v2f b2 = {wf5[2][kb][m16], wf5[2][kb + 1][m16]};
        const v2f b3 = {wf5[3][kb][m16], wf5[3][kb + 1][m16]};
        const v2f b4 = {wf5[4][kb][m16], wf5[4][kb + 1][m16]};
        nI  = wmma_f32(afrag[kc], b0, nI);
        nJ  = wmma_f32(afrag[kc], b1, nJ);
        nII = wmma_f32(afrag[kc], b2, nII);
        nJJ = wmma_f32(afrag[kc], b3, nJJ);
        nIJ = wmma_f32(afrag[kc], b4, nIJ);
      }
    }

    // ---- running window update: sums += new - outgoing; ring <- new ------
    // (ring slot being overwritten holds the plane leaving the 9-window)
#pragma unroll
    for (int r = 0; r < 8; ++r) {
      const int idx = (r + hi * 8) * TILE + m16;
      const float o0 = ring[slot][0][idx];
      const float o1 = ring[slot][1][idx];
      const float o2 = ring[slot][2][idx];
      const float o3 = ring[slot][3][idx];
      const float o4 = ring[slot][4][idx];
      sI[r]  += nI[r]  - o0;
      sJ[r]  += nJ[r]  - o1;
      sII[r] += nII[r] - o2;
      sJJ[r] += nJJ[r] - o3;
      sIJ[r] += nIJ[r] - o4;
      ring[slot][0][idx] = nI[r];
      ring[slot][1][idx] = nJ[r];
      ring[slot][2][idx] = nII[r];
      ring[slot][3][idx] = nJJ[r];
      ring[slot][4][idx] = nIJ[r];
    }

    // ---- full 9-window resident: emit output plane d = d0 + zp - 8 -------
    // combine straight from registers (WMMA C/D layout covers all 256 cells)
    if (zp >= 2 * HALO) {
#pragma unroll
      for (int r = 0; r < 8; ++r) {
        const float Is = sI[r],  Js = sJ[r];
        const float I2 = sII[r], J2 = sJJ[r], IJ = sIJ[r];
        const float uI = Is * (1.0f / WIN_SZ);
        const float uJ = Js * (1.0f / WIN_SZ);
        const float cross = IJ - uJ * Is - uI * Js + uI * uJ * WIN_SZ;
        const float Iv = I2 - 2.0f * uI * Is + uI * uI * WIN_SZ;
        const float Jv = J2 - 2.0f * uJ * Js + uJ * uJ * WIN_SZ;
        cc_acc += cross * cross / (Iv * Jv + 1e-5f);
      }
    }
  }

  // wave32 reduction + one atomic per wave
  for (int o = 16; o; o >>= 1) cc_acc += __shfl_down(cc_acc, o);
  if (lane == 0) atomicAdd(&ws[0], cc_acc);
}

__global__ void grad_kernel(const float* __restrict__ flow,
                            float* __restrict__ ws) {
  const int HW = H_ * W_;
  const int N = 3 * D_ * HW;
  float sdy = 0.f, sdx = 0.f, sdz = 0.f;
  for (int i = blockIdx.x * blockDim.x + threadIdx.x; i < N;
       i += gridDim.x * blockDim.x) {
    const int w = i % W_;
    const int t = i / W_;
    const int h = t % H_;
    const int d = (t / H_) % D_;
    const float v = flow[i];
    if (d + 1 < D_) { const float df = flow[i + HW] - v; sdy += df * df; }
    if (h + 1 < H_) { const float df = flow[i + W_] - v; sdx += df * df; }
    if (w + 1 < W_) { const float df = flow[i + 1]  - v; sdz += df * df; }
  }
  for (int o = 16; o; o >>= 1) {
    sdy += __shfl_down(sdy, o);
    sdx += __shfl_down(sdx, o);
    sdz += __shfl_down(sdz, o);
  }
  if ((threadIdx.x & 31) == 0) {
    atomicAdd(&ws[1], sdy);
    atomicAdd(&ws[2], sdx);
    atomicAdd(&ws[3], sdz);
  }
}

__global__ void kl_kernel(const float* __restrict__ x,   // log-probs input
                          const float* __restrict__ t,   // target probs
                          float* __restrict__ ws) {
  const int N4 = (4 * D_ * H_ * W_) / 4;                 // divisible by 4
  const float4* __restrict__ x4 = (const float4*)x;
  const float4* __restrict__ t4 = (const float4*)t;
  float s = 0.f;
  for (int i = blockIdx.x * blockDim.x + threadIdx.x; i < N4;
       i += gridDim.x * blockDim.x) {
    const float4 tv = t4[i];
    const float4 xv = x4[i];
    s += tv.x * (logf(tv.x) - xv.x);
    s += tv.y * (logf(tv.y) - xv.y);
    s += tv.z * (logf(tv.z) - xv.z);
    s += tv.w * (logf(tv.w) - xv.w);
  }
  for (int o = 16; o; o >>= 1) s += __shfl_down(s, o);
  if ((threadIdx.x & 31) == 0) atomicAdd(&ws[4], s);
}

__global__ void zero_ws_kernel(float* ws) {
  if (threadIdx.x < 8) ws[threadIdx.x] = 0.0f;
}

__global__ void finalize_kernel(const float* __restrict__ ws,
                                float* __restrict__ out) {
  if (threadIdx.x == 0 && blockIdx.x == 0) {
    const float NV = (float)D_ * (float)H_ * (float)W_;
    const float ncc = -ws[0] / NV;
    const float mdy = ws[1] / (3.0f * (D_ - 1) * H_ * W_);
    const float mdx = ws[2] / (3.0f * D_ * (H_ - 1) * W_);
    const float mdz = ws[3] / (3.0f * D_ * H_ * (W_ - 1));
    const float grad = (mdx + mdy + mdz) * (1.0f / 3.0f);
    const float kl = ws[4] / (4.0f * NV);
    out[0] = 10.0f * ncc + 15.0f * grad + 20.0f * kl;
    out[1] = ncc;
    out[2] = grad;
    out[3] = kl;
  }
}

extern "C" void kernel_launch(void* const* d_in, const int* in_sizes, int n_in,
                              void* d_out, int out_size, void* d_ws,
                              size_t ws_size, hipStream_t stream) {
  const float* y    = (const float*)d_in[0];  // (1,1,D,H,W)
  const float* tgt  = (const float*)d_in[1];  // (1,1,D,H,W)
  const float* flow = (const float*)d_in[2];  // (1,3,D,H,W)
  const float* wx   = (const float*)d_in[3];  // (1,4,D,H,W) log-probs
  const float* tl   = (const float*)d_in[4];  // (1,4,D,H,W) probs
  float* out = (float*)d_out;
  float* ws  = (float*)d_ws;

  zero_ws_kernel<<<1, 32, 0, stream>>>(ws);
  // 12 (H tiles) * 10 (W tiles) * 8 (D segments) = 960 single-wave WGs
  ncc_kernel<<<(H_ / TILE) * (W_ / TILE) * SEGS, 32, 0, stream>>>(y, tgt, ws);
  grad_kernel<<<2048, 256, 0, stream>>>(flow, ws);
  kl_kernel<<<2048, 256, 0, stream>>>(wx, tl, ws);
  finalize_kernel<<<1, 32, 0, stream>>>(ws, out);
}